// PositionalEncoding_62938450756192
// MI455X (gfx1250) — compile-verified
//
#include <hip/hip_runtime.h>
#include <stdint.h>

// Problem constants (b=8, c=256, h=w=256, k=_MP[256]=128)
#define HW        65536               // h*w elements per (b,c) row
#define K_SEL     128                 // top-k
#define NTHREADS  512                 // 16 waves (wave32)
#define NWAVES    (NTHREADS / 32)
#define CHUNK     (NTHREADS * 4)      // 2048 floats per streamed chunk (b128/thread)
#define NCHUNK    (HW / CHUNK)        // 32 chunks per row
#define NBINS     65536               // 16-bit radix histogram (256KB of the 320KB WGP LDS)
#define BPT       (NBINS / NTHREADS)  // 128 bins owned per thread

// Order-preserving float->uint key: larger float => larger key.
__device__ __forceinline__ uint32_t fkey(float f) {
  uint32_t b = __float_as_uint(f);
  uint32_t m = (uint32_t)((int32_t)b >> 31);
  return b ^ (m | 0x80000000u);
}

// CDNA5 async global->LDS load (GVS mode): SADDR 64b sgpr base + VADDR 32b vgpr
// byte offset; VDST = LDS byte address. Each thread owns a private 16B slot.
__device__ __forceinline__ void async_issue(uint64_t gbase, int chunk, float* stagebuf, int t) {
  uint32_t lds_addr = (uint32_t)(size_t)(stagebuf + t * 4);
  uint32_t goff     = (uint32_t)(chunk * CHUNK + t * 4) * 4u;
  asm volatile("global_load_async_to_lds_b128 %0, %1, %2"
               :: "v"(lds_addr), "v"(goff), "s"(gbase)
               : "memory");
}
__device__ __forceinline__ void wait_async_le1() {
  asm volatile("s_wait_asynccnt 0x1" ::: "memory");
}
__device__ __forceinline__ void wait_async_0() {
  asm volatile("s_wait_asynccnt 0x0" ::: "memory");
}

// Intra-wave inclusive scan (wave32, no barriers — cross-lane VALU only).
__device__ __forceinline__ uint32_t wave_incl_scan(uint32_t v, int lane) {
  #pragma unroll
  for (int off = 1; off < 32; off <<= 1) {
    uint32_t n = __shfl_up(v, (unsigned)off, 32);
    if (lane >= off) v += n;
  }
  return v;
}

// Workgroup inclusive scan of one value per thread. wls = NWAVES-entry LDS
// scratch. Returns inclusive prefix; *total = grand total. 3 barriers.
__device__ __forceinline__ uint32_t wg_incl_scan_val(uint32_t v, uint32_t* wls, int t,
                                                     uint32_t* total) {
  const int lane = t & 31, wave = t >> 5;
  uint32_t ws = wave_incl_scan(v, lane);
  if (lane == 31) wls[wave] = ws;
  __syncthreads();
  if (wave == 0) {
    uint32_t wv = (lane < NWAVES) ? wls[lane] : 0u;
    uint32_t wsc = wave_incl_scan(wv, lane);
    if (lane < NWAVES) wls[lane] = wsc;
  }
  __syncthreads();
  uint32_t base = (wave > 0) ? wls[wave - 1] : 0u;
  uint32_t tot  = wls[NWAVES - 1];
  __syncthreads();                        // protect wls before next reuse
  *total = tot;
  return base + ws;
}

// From the NBINS histogram, find bin T and count-above such that
// (#elems in bins > T) < target <= (#elems in bins >= T).
__device__ __forceinline__ void find_threshold(uint32_t* hist, uint32_t* wls,
                                               uint32_t* vBin, uint32_t* vAbove,
                                               uint32_t target, int t) {
  const int base = t * BPT;
  uint32_t s = 0;
  #pragma unroll 4
  for (int j = 0; j < BPT; ++j) s += hist[base + j];
  uint32_t total;
  uint32_t incl = wg_incl_scan_val(s, wls, t, &total);
  uint32_t run = total - incl;            // elems in bins owned by threads > t
  for (int j = BPT - 1; j >= 0; --j) {
    uint32_t c = hist[base + j];
    if (run < target && run + c >= target) { *vBin = (uint32_t)(base + j); *vAbove = run; }
    run += c;
  }
  __syncthreads();
}

extern "C" __global__ __launch_bounds__(NTHREADS)
void topk_posenc_kernel(const float* __restrict__ x,
                        const float* __restrict__ pe,
                        float* __restrict__ out) {
  extern __shared__ uint32_t smem[];
  uint32_t* hist  = smem;                               // NBINS u32 (256 KB)
  float*    stage = (float*)(smem + NBINS);             // 2 * NTHREADS * 4 floats (16 KB)
  uint32_t* wls   = smem + NBINS + 2 * NTHREADS * 4;    // NWAVES u32 scan scratch
  uint32_t* vars  = wls + NWAVES;                       // 8 u32

  const int t = threadIdx.x;
  const int row = blockIdx.x;
  const uint64_t rowbase = (uint64_t)(x + (size_t)row * HW);
  float* orow = out + (size_t)row * K_SEL;

  // ---------------- Pass 1: histogram of key>>16 ----------------
  for (int j = t; j < NBINS; j += NTHREADS) hist[j] = 0u;
  __syncthreads();

  async_issue(rowbase, 0, stage, t);
  for (int c = 0; c < NCHUNK; ++c) {
    float* buf = stage + (c & 1) * (NTHREADS * 4);
    if (c + 1 < NCHUNK) { async_issue(rowbase, c + 1, stage + ((c + 1) & 1) * (NTHREADS * 4), t); wait_async_le1(); }
    else                { wait_async_0(); }
    float4 v4 = *reinterpret_cast<const float4*>(buf + t * 4);
    atomicAdd(&hist[fkey(v4.x) >> 16], 1u);
    atomicAdd(&hist[fkey(v4.y) >> 16], 1u);
    atomicAdd(&hist[fkey(v4.z) >> 16], 1u);
    atomicAdd(&hist[fkey(v4.w) >> 16], 1u);
  }
  __syncthreads();

  find_threshold(hist, wls, &vars[0], &vars[1], K_SEL, t);
  const uint32_t T1       = vars[0];
  const uint32_t cntAbove = vars[1];
  const uint32_t need     = K_SEL - cntAbove;

  // ---------------- Pass 2: low 16 bits of tie-prefix elements ----------------
  for (int j = t; j < NBINS; j += NTHREADS) hist[j] = 0u;
  __syncthreads();

  async_issue(rowbase, 0, stage, t);
  for (int c = 0; c < NCHUNK; ++c) {
    float* buf = stage + (c & 1) * (NTHREADS * 4);
    if (c + 1 < NCHUNK) { async_issue(rowbase, c + 1, stage + ((c + 1) & 1) * (NTHREADS * 4), t); wait_async_le1(); }
    else                { wait_async_0(); }
    float4 v4 = *reinterpret_cast<const float4*>(buf + t * 4);
    float va[4] = {v4.x, v4.y, v4.z, v4.w};
    #pragma unroll
    for (int j = 0; j < 4; ++j) {
      uint32_t u = fkey(va[j]);
      if ((u >> 16) == T1) atomicAdd(&hist[u & 0xFFFFu], 1u);
    }
  }
  __syncthreads();

  find_threshold(hist, wls, &vars[2], &vars[3], need, t);
  const uint32_t T2          = vars[2];
  const uint32_t cntAboveLow = vars[3];
  const uint32_t uThr  = (T1 << 16) | T2;     // exact rank-k threshold key
  const uint32_t need2 = need - cntAboveLow;  // exact-equal elems to take (lowest index first)

  // ---------------- Pass 3: ordered selection, one fused packed scan per chunk ----
  if (t == 0) { vars[4] = 0u; vars[5] = 0u; } // gSeen, eqSeen (global running counts)
  __syncthreads();

  async_issue(rowbase, 0, stage, t);
  for (int c = 0; c < NCHUNK; ++c) {
    float* buf = stage + (c & 1) * (NTHREADS * 4);
    if (c + 1 < NCHUNK) { async_issue(rowbase, c + 1, stage + ((c + 1) & 1) * (NTHREADS * 4), t); wait_async_le1(); }
    else                { wait_async_0(); }
    float4 v4 = *reinterpret_cast<const float4*>(buf + t * 4);
    float va[4] = {v4.x, v4.y, v4.z, v4.w};

    uint32_t g[4], e[4], gsum = 0, esum = 0;
    #pragma unroll
    for (int j = 0; j < 4; ++j) {
      uint32_t u = fkey(va[j]);
      g[j] = (u > uThr);
      e[j] = (u == uThr);
      gsum += g[j];
      esum += e[j];
    }
    // One packed scan: high half = greater-count, low half = equal-count
    // (chunk totals <= 2048 each, so 16-bit fields cannot overflow).
    uint32_t total;
    uint32_t incl = wg_incl_scan_val((gsum << 16) | esum, wls, t, &total);
    uint32_t excl = incl - ((gsum << 16) | esum);
    uint32_t gBefore = vars[4] + (excl >> 16);
    uint32_t eBefore = vars[5] + (excl & 0xFFFFu);

    const int base = c * CHUNK + t * 4;
    uint32_t gRun = 0, eRun = 0;
    #pragma unroll
    for (int j = 0; j < 4; ++j) {
      uint32_t eqRank = eBefore + eRun;
      if (g[j] | (e[j] & (uint32_t)(eqRank < need2))) {
        // slot = (#greater before) + (#selected-equal before); selected-equal
        // before = min(eqRank, need2), and for a selected equal eqRank < need2.
        uint32_t eqSelBefore = eqRank < need2 ? eqRank : need2;
        uint32_t slot = (gBefore + gRun) + eqSelBefore;
        int idx = base + j;                 // pos_enc[idx/256, idx%256] == pe_flat[idx]
        orow[slot] = pe[idx] + va[j];
      }
      gRun += g[j];
      eRun += e[j];
    }
    __syncthreads();
    if (t == 0) { vars[4] += total >> 16; vars[5] += total & 0xFFFFu; }
    __syncthreads();
  }
}

extern "C" void kernel_launch(void* const* d_in, const int* in_sizes, int n_in,
                              void* d_out, int out_size, void* d_ws, size_t ws_size,
                              hipStream_t stream) {
  const float* x  = (const float*)d_in[0];   // [b, c, h, w] fp32
  const float* pe = (const float*)d_in[1];   // [256, 256] fp32 (flat lookup)
  float* out      = (float*)d_out;           // [b, c, 128] fp32

  const int rows = in_sizes[0] / HW;         // b*c = 2048
  const size_t smem_bytes = (size_t)(NBINS + 2 * NTHREADS * 4 + NWAVES + 8) * sizeof(uint32_t);

  topk_posenc_kernel<<<dim3(rows), dim3(NTHREADS), smem_bytes, stream>>>(x, pe, out);
}